// DRNLFilterbank_56092272886026
// MI455X (gfx1250) — compile-verified
//
#include <hip/hip_runtime.h>

typedef __attribute__((ext_vector_type(2))) float v2f;
typedef __attribute__((ext_vector_type(8))) float v8f;

#define NFILT 50
#define BATCH 8
#define TLEN  16000
#define BLKT  16
#define NBLK  (TLEN / BLKT)

// ---------------------------------------------------------------------------
// WMMA wrapper: D(16x16 f32) = A(16x4 f32) * B(4x16 f32) + C
// A operand: lane m=lane&15 is row M; VGPR0 = K{0,2}(lo/hi half-wave), VGPR1 = K{1,3}
// B operand: lane n=lane&15 is col N; VGPR0 = K{0,2}, VGPR1 = K{1,3}
// C/D: lanes 0-15 M=r, lanes 16-31 M=r+8, N = lane&15
// ---------------------------------------------------------------------------
__device__ __forceinline__ v8f wmma_f32(v2f a, v2f b, v8f c) {
  return __builtin_amdgcn_wmma_f32_16x16x4_f32(false, a, false, b, (short)0, c,
                                               false, false);
}

// ===========================================================================
// Precompute kernel: build block state-space matrices for each (filter,
// cascade).  Cascade of K transposed-DF2 biquads == LTI system of order 2K.
// Over a 16-sample block:  Y = H*X + S*Z0 ;  Zend = G*X + A*Z0.
// Probe p<16:  X=e_p, Z0=0  -> column p of H (from Y) and G (from Zend)
// Probe p>=16: X=0, Z0=e_{p-16} -> column of S and A (zero beyond 2K rows).
// Stored row-major 16x32: Py = [H | S], Pz = [G | A].
// ===========================================================================
__global__ void drnl_precompute(const float* __restrict__ sos_lin,
                                const float* __restrict__ sos_nl1,
                                const float* __restrict__ sos_nl2,
                                float* __restrict__ Py, float* __restrict__ Pz) {
  const int f = blockIdx.x;   // filter 0..49
  const int c = blockIdx.y;   // cascade 0=lin(6), 1=nl1(3), 2=nl2(6)
  const int p = threadIdx.x;  // probe 0..31

  const float* sos;
  int K;
  if (c == 0)      { sos = sos_lin + f * 6 * 6; K = 6; }
  else if (c == 1) { sos = sos_nl1 + f * 3 * 6; K = 3; }
  else             { sos = sos_nl2 + f * 6 * 6; K = 6; }

  float b0[6], b1[6], b2[6], a1[6], a2[6];
  for (int k = 0; k < K; k++) {
    b0[k] = sos[k * 6 + 0];
    b1[k] = sos[k * 6 + 1];
    b2[k] = sos[k * 6 + 2];
    a1[k] = sos[k * 6 + 4];
    a2[k] = sos[k * 6 + 5];
  }

  float z[6][2];
  for (int k = 0; k < 6; k++) { z[k][0] = 0.f; z[k][1] = 0.f; }
  if (p >= 16) {
    int j = p - 16;
    if (j < 2 * K) z[j >> 1][j & 1] = 1.0f;
  }

  float* py = Py + (size_t)(f * 3 + c) * 16 * 32;
  float* pz = Pz + (size_t)(f * 3 + c) * 16 * 32;

  for (int t = 0; t < BLKT; t++) {
    float u = (p < 16 && t == p) ? 1.0f : 0.0f;
    for (int k = 0; k < K; k++) {
      float y  = b0[k] * u + z[k][0];
      z[k][0]  = b1[k] * u - a1[k] * y + z[k][1];
      z[k][1]  = b2[k] * u - a2[k] * y;
      u = y;
    }
    py[t * 32 + p] = u;  // column p, row t of [H|S]
  }
  for (int r = 0; r < 16; r++) {
    float v = (r < 2 * K) ? z[r >> 1][r & 1] : 0.0f;
    pz[r * 32 + p] = v;  // column p, row r of [G|A]
  }
}

// ---------------------------------------------------------------------------
// C-layout (v8f accumulator) -> B-layout (4x v2f K-slices) via a 1KB LDS tile.
// Single-wave workgroup: barriers lower to S_NOP; LDS RAW handled by dscnt.
// ---------------------------------------------------------------------------
__device__ __forceinline__ void c_to_b(float (*xch)[17], const v8f& cacc,
                                       v2f zb[4], int m, int hi) {
  __syncthreads();
#pragma unroll
  for (int r = 0; r < 8; r++) xch[r + 8 * hi][m] = cacc[r];
  __syncthreads();
#pragma unroll
  for (int s = 0; s < 4; s++) {
    const int k0 = 4 * s + 2 * hi;
    v2f t;
    t.x = xch[k0][m];
    t.y = xch[k0 + 1][m];
    zb[s] = t;
  }
}

// ===========================================================================
// Main kernel: one wave per filter.  Per 16-sample block, three block
// state-space cascades (16 WMMAs each) + pointwise compression.
// Columns 0-7 of the WMMA N-dimension carry the 8 batch sequences; columns
// 8-15 redundantly recompute columns 0-7 (clamped load address) so the loop
// body is completely branch-free — only the final store is predicated.
// ===========================================================================
__global__ __launch_bounds__(32) void drnl_block(
    const float* __restrict__ x, const float* __restrict__ g_lin,
    const float* __restrict__ a_nl, const float* __restrict__ b_nl,
    const float* __restrict__ Py, const float* __restrict__ Pz,
    float* __restrict__ out) {
  __shared__ float xch[16][17];

  const int f    = blockIdx.x;
  const int lane = threadIdx.x & 31;
  const int m    = lane & 15;   // A-row / B-col / C-col index
  const int hi   = lane >> 4;   // half-wave select
  const int nb   = m & 7;       // clamped batch index (cols 8-15 duplicate 0-7)

  // ---- load per-filter block matrices as WMMA A-operands (register resident)
  v2f Ha[3][4], Sa[3][4], Ga[3][4], Aa[3][4];
#pragma unroll
  for (int c = 0; c < 3; c++) {
    const float* py = Py + (size_t)(f * 3 + c) * 16 * 32 + m * 32;
    const float* pz = Pz + (size_t)(f * 3 + c) * 16 * 32 + m * 32;
#pragma unroll
    for (int s = 0; s < 4; s++) {
      const int k0 = 4 * s + 2 * hi;
      v2f t;
      t.x = py[k0];      t.y = py[k0 + 1];      Ha[c][s] = t;
      t.x = py[16 + k0]; t.y = py[16 + k0 + 1]; Sa[c][s] = t;
      t.x = pz[k0];      t.y = pz[k0 + 1];      Ga[c][s] = t;
      t.x = pz[16 + k0]; t.y = pz[16 + k0 + 1]; Aa[c][s] = t;
    }
  }
  const float g    = g_lin[f];
  const float A_nl = a_nl[f];
  const float B_nl = b_nl[f];

  v2f Zl[4], Z1[4], Z2[4];
#pragma unroll
  for (int s = 0; s < 4; s++) {
    v2f z; z.x = 0.f; z.y = 0.f;
    Zl[s] = z; Z1[s] = z; Z2[s] = z;
  }

  const float* xp = x + (size_t)nb * TLEN;                  // always valid
  float*       op = out + ((size_t)m * NFILT + f) * TLEN;   // valid iff m<8

  for (int blk = 0; blk < NBLK; ++blk) {
    const int t0 = blk * BLKT;

    // ---- gather X block into B-layout (branch-free, cols 8-15 duplicate) --
    v2f Xr[4];
#pragma unroll
    for (int s = 0; s < 4; s++) {
      const int k0 = 4 * s + 2 * hi;
      const float2 t = *reinterpret_cast<const float2*>(xp + t0 + k0);
      v2f xv; xv.x = t.x; xv.y = t.y;
      Xr[s] = xv;
    }
    __builtin_prefetch(xp + t0 + BLKT, 0, 1);

    // ================= linear path: 6-biquad cascade =================
    v2f Xl[4];
#pragma unroll
    for (int s = 0; s < 4; s++) Xl[s] = Xr[s] * g;

    v8f Ylin = {0.f, 0.f, 0.f, 0.f, 0.f, 0.f, 0.f, 0.f};
#pragma unroll
    for (int s = 0; s < 4; s++) {
      Ylin = wmma_f32(Ha[0][s], Xl[s], Ylin);
      Ylin = wmma_f32(Sa[0][s], Zl[s], Ylin);
    }
    v8f Zacc = {0.f, 0.f, 0.f, 0.f, 0.f, 0.f, 0.f, 0.f};
#pragma unroll
    for (int s = 0; s < 4; s++) {
      Zacc = wmma_f32(Ga[0][s], Xl[s], Zacc);
      Zacc = wmma_f32(Aa[0][s], Zl[s], Zacc);
    }
    c_to_b(xch, Zacc, Zl, m, hi);

    // ================= NL path, cascade 1 (3 biquads) ================
    v8f Y1 = {0.f, 0.f, 0.f, 0.f, 0.f, 0.f, 0.f, 0.f};
#pragma unroll
    for (int s = 0; s < 4; s++) {
      Y1 = wmma_f32(Ha[1][s], Xr[s], Y1);
      Y1 = wmma_f32(Sa[1][s], Z1[s], Y1);
    }
    v8f Zacc1 = {0.f, 0.f, 0.f, 0.f, 0.f, 0.f, 0.f, 0.f};
#pragma unroll
    for (int s = 0; s < 4; s++) {
      Zacc1 = wmma_f32(Ga[1][s], Xr[s], Zacc1);
      Zacc1 = wmma_f32(Aa[1][s], Z1[s], Zacc1);
    }
    c_to_b(xch, Zacc1, Z1, m, hi);

    // ---- broken-stick compression: sign(y)*min(a*|y|, b*|y|^0.25) ----
    v8f Yc;
#pragma unroll
    for (int r = 0; r < 8; r++) {
      float v  = Y1[r];
      float av = fmaxf(fabsf(v), 1e-12f);
      float lo = A_nl * av;
      float up = B_nl * sqrtf(sqrtf(av));
      float mn = fminf(lo, up);
      float sg = (v > 0.f) ? 1.f : ((v < 0.f) ? -1.f : 0.f);
      Yc[r] = sg * mn;
    }
    v2f Cb[4];
    c_to_b(xch, Yc, Cb, m, hi);

    // ================= NL path, cascade 2 (6 biquads) ================
    v8f Y2 = {0.f, 0.f, 0.f, 0.f, 0.f, 0.f, 0.f, 0.f};
#pragma unroll
    for (int s = 0; s < 4; s++) {
      Y2 = wmma_f32(Ha[2][s], Cb[s], Y2);
      Y2 = wmma_f32(Sa[2][s], Z2[s], Y2);
    }
    v8f Zacc2 = {0.f, 0.f, 0.f, 0.f, 0.f, 0.f, 0.f, 0.f};
#pragma unroll
    for (int s = 0; s < 4; s++) {
      Zacc2 = wmma_f32(Ga[2][s], Cb[s], Zacc2);
      Zacc2 = wmma_f32(Aa[2][s], Z2[s], Zacc2);
    }
    c_to_b(xch, Zacc2, Z2, m, hi);

    // ---- y = y_lin + y_nl ; C-layout rows are consecutive t -> B128 stores
    v8f Yo = Ylin + Y2;
    if (m < BATCH) {
      float* o = op + t0 + 8 * hi;
      float4 p0 = make_float4(Yo[0], Yo[1], Yo[2], Yo[3]);
      float4 p1 = make_float4(Yo[4], Yo[5], Yo[6], Yo[7]);
      reinterpret_cast<float4*>(o)[0] = p0;
      reinterpret_cast<float4*>(o)[1] = p1;
    }
  }
}

// ===========================================================================
extern "C" void kernel_launch(void* const* d_in, const int* in_sizes, int n_in,
                              void* d_out, int out_size, void* d_ws,
                              size_t ws_size, hipStream_t stream) {
  (void)in_sizes; (void)n_in; (void)out_size; (void)ws_size;
  const float* x       = (const float*)d_in[0];
  const float* g_lin   = (const float*)d_in[1];
  const float* a_nl    = (const float*)d_in[2];
  const float* b_nl    = (const float*)d_in[3];
  const float* sos_lin = (const float*)d_in[4];
  const float* sos_nl1 = (const float*)d_in[5];
  const float* sos_nl2 = (const float*)d_in[6];

  float* Py = (float*)d_ws;                       // 50*3*16*32 floats
  float* Pz = Py + (size_t)NFILT * 3 * 16 * 32;   // same size

  dim3 gpre(NFILT, 3);
  drnl_precompute<<<gpre, 32, 0, stream>>>(sos_lin, sos_nl1, sos_nl2, Py, Pz);

  drnl_block<<<NFILT, 32, 0, stream>>>(x, g_lin, a_nl, b_nl, Py, Pz,
                                       (float*)d_out);
}